// GAT_16698832847058
// MI455X (gfx1250) — compile-verified
//
#include <hip/hip_runtime.h>
#include <math.h>

#define NFEAT 128
#define NHID 32
#define NHEADS 4
#define NCLASS 32
#define GAT_ALPHA 0.2f

typedef __attribute__((ext_vector_type(2))) float v2f;
typedef __attribute__((ext_vector_type(8))) float v8f;

// ---------------------------------------------------------------------------
// ordered float <-> uint key for atomic segment-max (monotone mapping)
// ---------------------------------------------------------------------------
__device__ __forceinline__ unsigned floatKey(float f) {
  unsigned u = __float_as_uint(f);
  return u ^ ((u >> 31) ? 0xFFFFFFFFu : 0x80000000u);
}
__device__ __forceinline__ float keyFloat(unsigned k) {
  unsigned u = k ^ ((k >> 31) ? 0x80000000u : 0xFFFFFFFFu);
  return __uint_as_float(u);
}

// ---------------------------------------------------------------------------
// f32 WMMA GEMM: C[M,Nc] = A[M,K] @ B[K,Nc], row-major, M%16==0, Nc%16==0,
// K%4==0. One wave computes one 16x16 tile via V_WMMA_F32_16X16X4_F32.
// A-frag (16x4): lane l holds row M=l%16; vgpr v holds K = 2*(l/16)+v.
// C/D (16x16): vgpr r, lane l -> row = r + 8*(l/16), col = l%16.
// ---------------------------------------------------------------------------
__global__ void gemm_f32_wmma(const float* __restrict__ A,
                              const float* __restrict__ B,
                              float* __restrict__ C,
                              int M, int Nc, int K) {
  const int gid  = blockIdx.x * blockDim.x + threadIdx.x;
  const int wave = gid >> 5;
  const int lane = gid & 31;
  const int tilesM = M >> 4;
  const int tilesN = Nc >> 4;
  if (wave >= tilesM * tilesN) return;          // uniform per-wave exit
  const int tm = wave % tilesM;
  const int tn = wave / tilesM;

  const int m  = (tm << 4) + (lane & 15);
  const int n  = (tn << 4) + (lane & 15);
  const int kh = (lane >> 4) << 1;              // 0 or 2

  v8f acc = {};
  const float* Arow = A + (size_t)m * K;
  for (int k0 = 0; k0 < K; k0 += 4) {
    v2f a, b;
    a.x = Arow[k0 + kh + 0];
    a.y = Arow[k0 + kh + 1];
    b.x = B[(size_t)(k0 + kh + 0) * Nc + n];
    b.y = B[(size_t)(k0 + kh + 1) * Nc + n];
    acc = __builtin_amdgcn_wmma_f32_16x16x4_f32(
        /*neg_a=*/false, a, /*neg_b=*/false, b,
        /*c_mod=*/(short)0, acc, /*reuse_a=*/false, /*reuse_b=*/false);
  }

  const int row0 = (tm << 4) + ((lane >> 4) << 3);
  const int coln = (tn << 4) + (lane & 15);
#pragma unroll
  for (int r = 0; r < 8; ++r)
    C[(size_t)(row0 + r) * Nc + coln] = acc[r];
}

// ---------------------------------------------------------------------------
// weight repacks
// ---------------------------------------------------------------------------
__global__ void pack_wcat(const float* __restrict__ W_heads, float* __restrict__ Wcat) {
  int idx = blockIdx.x * blockDim.x + threadIdx.x;
  if (idx >= NFEAT * NFEAT) return;
  int k = idx / NFEAT, c = idx % NFEAT;
  int h = c / NHID, j = c % NHID;
  Wcat[idx] = W_heads[(size_t)h * NFEAT * NHID + (size_t)k * NHID + j];
}
__global__ void pack_wrest(const float* __restrict__ res_W, float* __restrict__ WT) {
  int idx = blockIdx.x * blockDim.x + threadIdx.x;
  if (idx >= NFEAT * NCLASS) return;
  int k = idx / NCLASS, j = idx % NCLASS;
  WT[idx] = res_W[(size_t)j * NFEAT + k];    // res_W is [NCLASS, NFEAT]
}

// ---------------------------------------------------------------------------
// attention coefficients
// ---------------------------------------------------------------------------
__global__ void head_coef(const float* __restrict__ H, const float* __restrict__ a,
                          float* __restrict__ esrc, float* __restrict__ edst, int N) {
  int t = blockIdx.x * blockDim.x + threadIdx.x;
  if (t >= N * NHEADS) return;
  int n = t / NHEADS, h = t % NHEADS;
  const float* hr = H + (size_t)n * NFEAT + h * NHID;
  const float* av = a + h * 2 * NHID;        // a_heads [NHEADS, 2*NHID, 1]
  float s = 0.f, d = 0.f;
#pragma unroll 8
  for (int j = 0; j < NHID; ++j) {
    float v = hr[j];
    s += v * av[j];
    d += v * av[NHID + j];
  }
  esrc[t] = s;   // layout [n*NHEADS + h]
  edst[t] = d;
}

__global__ void out_coef(const float* __restrict__ O1, const float* __restrict__ a,
                         float* __restrict__ esrc2, float* __restrict__ edst2, int N) {
  int n = blockIdx.x * blockDim.x + threadIdx.x;
  if (n >= N) return;
  const float* o = O1 + (size_t)n * NCLASS;
  float s = 0.f, d = 0.f;
#pragma unroll 8
  for (int j = 0; j < NCLASS; ++j) {
    float v = o[j];
    s += v * a[j];
    d += v * a[NCLASS + j];
  }
  esrc2[n] = s;
  edst2[n] = d;
}

// ---------------------------------------------------------------------------
// edge passes (generic in nh; esrc/edst layout [node*nh + h], ebuf [h*E + e],
// emax/denom [h*N + node])
// ---------------------------------------------------------------------------
__global__ void edge_logits(const int* __restrict__ row, const int* __restrict__ col,
                            const float* __restrict__ esrc, const float* __restrict__ edst,
                            float* __restrict__ ebuf, unsigned* __restrict__ emax,
                            int E, int N, int nh) {
  int i = blockIdx.x * blockDim.x + threadIdx.x;
  if (i >= E) return;
  int r = row[i], c = col[i];
  for (int h = 0; h < nh; ++h) {
    float e = esrc[(size_t)r * nh + h] + edst[(size_t)c * nh + h];
    e = (e > 0.f) ? e : GAT_ALPHA * e;       // leaky relu
    ebuf[(size_t)h * E + i] = e;
    atomicMax(&emax[(size_t)h * N + r], floatKey(e));
  }
}

__global__ void edge_expsum(const int* __restrict__ row,
                            float* __restrict__ ebuf, const unsigned* __restrict__ emax,
                            float* __restrict__ denom, int E, int N, int nh) {
  int i = blockIdx.x * blockDim.x + threadIdx.x;
  if (i >= E) return;
  int r = row[i];
  for (int h = 0; h < nh; ++h) {
    float e = ebuf[(size_t)h * E + i];
    float m = keyFloat(emax[(size_t)h * N + r]);
    float ex = __expf(e - m);
    ebuf[(size_t)h * E + i] = ex;
    atomicAdd(&denom[(size_t)h * N + r], ex);
  }
}

// wave-per-edge, 128 features (4 heads x 32): lane l owns feats 4l..4l+3, head l>>3
__global__ void edge_agg_heads(const int* __restrict__ row, const int* __restrict__ col,
                               const float* __restrict__ ebuf, const float* __restrict__ denom,
                               const float* __restrict__ H, float* __restrict__ hprime,
                               int E, int N) {
  int g = blockIdx.x * blockDim.x + threadIdx.x;
  int e = g >> 5, lane = g & 31;
  if (e >= E) return;                         // uniform per-wave exit
  int r = row[e], c = col[e];
  int f = lane << 2;
  int h = lane >> 3;
  float att = ebuf[(size_t)h * E + e] / fmaxf(denom[(size_t)h * N + r], 1e-8f);
  const float4 hv = *(const float4*)(H + (size_t)c * NFEAT + f);
  float* dst = hprime + (size_t)r * NFEAT + f;
  atomicAdd(dst + 0, att * hv.x);
  atomicAdd(dst + 1, att * hv.y);
  atomicAdd(dst + 2, att * hv.z);
  atomicAdd(dst + 3, att * hv.w);
}

// wave-per-edge, 32 features, single "head"
__global__ void edge_agg_out(const int* __restrict__ row, const int* __restrict__ col,
                             const float* __restrict__ ebuf2, const float* __restrict__ denom2,
                             const float* __restrict__ O1, float* __restrict__ outp,
                             int E, int N) {
  int g = blockIdx.x * blockDim.x + threadIdx.x;
  int e = g >> 5, lane = g & 31;
  if (e >= E) return;
  int r = row[e], c = col[e];
  float att = ebuf2[e] / fmaxf(denom2[r], 1e-8f);
  atomicAdd(outp + (size_t)r * NCLASS + lane, att * O1[(size_t)c * NCLASS + lane]);
}

// ---------------------------------------------------------------------------
// elementwise ELU (in place)
// ---------------------------------------------------------------------------
__global__ void elu_k(float* __restrict__ a, int n) {
  int i = blockIdx.x * blockDim.x + threadIdx.x;
  if (i >= n) return;
  float v = a[i];
  a[i] = (v > 0.f) ? v : (__expf(v) - 1.f);
}

// ---------------------------------------------------------------------------
// relu + residual + bias + LayerNorm(32), wave-per-node (wave32, lane=channel)
// ---------------------------------------------------------------------------
__global__ void final_ln(const float* __restrict__ outp, const float* __restrict__ resid,
                         const float* __restrict__ res_b, const float* __restrict__ ln_g,
                         const float* __restrict__ ln_b, float* __restrict__ out, int N) {
  int g = blockIdx.x * blockDim.x + threadIdx.x;
  int n = g >> 5, c = g & 31;
  if (n >= N) return;
  float v = fmaxf(outp[(size_t)n * NCLASS + c], 0.f)
          + resid[(size_t)n * NCLASS + c] + res_b[c];
  float s = v;
#pragma unroll
  for (int o = 16; o > 0; o >>= 1) s += __shfl_xor(s, o, 32);
  float mean = s * (1.f / NCLASS);
  float d = v - mean;
  float q = d * d;
#pragma unroll
  for (int o = 16; o > 0; o >>= 1) q += __shfl_xor(q, o, 32);
  float var = q * (1.f / NCLASS);
  out[(size_t)n * NCLASS + c] = d * rsqrtf(var + 1e-5f) * ln_g[c] + ln_b[c];
}

// ---------------------------------------------------------------------------
// launcher
// ---------------------------------------------------------------------------
static inline int cdiv(long long a, long long b) { return (int)((a + b - 1) / b); }

extern "C" void kernel_launch(void* const* d_in, const int* in_sizes, int n_in,
                              void* d_out, int out_size, void* d_ws, size_t ws_size,
                              hipStream_t stream) {
  (void)n_in; (void)out_size; (void)ws_size;
  const float* x       = (const float*)d_in[0];
  const int*   edge    = (const int*)d_in[1];
  const float* W_heads = (const float*)d_in[2];
  const float* a_heads = (const float*)d_in[3];
  const float* W_out   = (const float*)d_in[4];
  const float* a_out   = (const float*)d_in[5];
  const float* res_W   = (const float*)d_in[6];
  const float* res_b   = (const float*)d_in[7];
  const float* ln_g    = (const float*)d_in[8];
  const float* ln_b    = (const float*)d_in[9];
  float* out = (float*)d_out;

  const int N = in_sizes[0] / NFEAT;
  const int E = in_sizes[1] / 2;
  const int* row = edge;
  const int* col = edge + E;

  // ---- carve workspace (256B aligned chunks) ----
  char* p = (char*)d_ws;
  auto carve = [&p](size_t bytes) {
    char* r = p;
    p += (bytes + 255) & ~(size_t)255;
    return (void*)r;
  };
  float*    H      = (float*)   carve(sizeof(float) * (size_t)N * NFEAT);
  float*    hprime = (float*)   carve(sizeof(float) * (size_t)N * NFEAT);
  float*    O1     = (float*)   carve(sizeof(float) * (size_t)N * NCLASS);
  float*    outp   = (float*)   carve(sizeof(float) * (size_t)N * NCLASS);
  float*    resid  = (float*)   carve(sizeof(float) * (size_t)N * NCLASS);
  float*    esrc   = (float*)   carve(sizeof(float) * (size_t)N * NHEADS);
  float*    edst   = (float*)   carve(sizeof(float) * (size_t)N * NHEADS);
  float*    ebuf   = (float*)   carve(sizeof(float) * (size_t)E * NHEADS);
  unsigned* emax   = (unsigned*)carve(sizeof(unsigned) * (size_t)N * NHEADS);
  float*    denom  = (float*)   carve(sizeof(float) * (size_t)N * NHEADS);
  float*    esrc2  = (float*)   carve(sizeof(float) * (size_t)N);
  float*    edst2  = (float*)   carve(sizeof(float) * (size_t)N);
  float*    ebuf2  = (float*)   carve(sizeof(float) * (size_t)E);
  unsigned* emax2  = (unsigned*)carve(sizeof(unsigned) * (size_t)N);
  float*    denom2 = (float*)   carve(sizeof(float) * (size_t)N);
  float*    Wcat   = (float*)   carve(sizeof(float) * NFEAT * NFEAT);
  float*    WresT  = (float*)   carve(sizeof(float) * NFEAT * NCLASS);

  const int T = 256;

  // zero accumulators (capture-safe memsets)
  hipMemsetAsync(hprime, 0, sizeof(float) * (size_t)N * NFEAT, stream);
  hipMemsetAsync(outp,   0, sizeof(float) * (size_t)N * NCLASS, stream);
  hipMemsetAsync(emax,   0, sizeof(unsigned) * (size_t)N * NHEADS, stream);
  hipMemsetAsync(denom,  0, sizeof(float) * (size_t)N * NHEADS, stream);
  hipMemsetAsync(emax2,  0, sizeof(unsigned) * (size_t)N, stream);
  hipMemsetAsync(denom2, 0, sizeof(float) * (size_t)N, stream);

  // weight repacks
  pack_wcat <<<cdiv(NFEAT * NFEAT, T), T, 0, stream>>>(W_heads, Wcat);
  pack_wrest<<<cdiv(NFEAT * NCLASS, T), T, 0, stream>>>(res_W, WresT);

  // H = x @ Wcat  [N,128]
  {
    long long waves = (long long)(N / 16) * (NFEAT / 16);
    gemm_f32_wmma<<<cdiv(waves * 32, T), T, 0, stream>>>(x, Wcat, H, N, NFEAT, NFEAT);
  }

  // per-head attention coefficients
  head_coef<<<cdiv((long long)N * NHEADS, T), T, 0, stream>>>(H, a_heads, esrc, edst, N);

  // multi-head edge softmax + aggregation (fused over 4 heads)
  edge_logits<<<cdiv(E, T), T, 0, stream>>>(row, col, esrc, edst, ebuf, emax, E, N, NHEADS);
  edge_expsum<<<cdiv(E, T), T, 0, stream>>>(row, ebuf, emax, denom, E, N, NHEADS);
  edge_agg_heads<<<cdiv((long long)E * 32, T), T, 0, stream>>>(row, col, ebuf, denom, H, hprime, E, N);

  // elu -> hcat (in place)
  elu_k<<<cdiv((long long)N * NFEAT, T), T, 0, stream>>>(hprime, N * NFEAT);

  // O1 = hcat @ W_out  [N,32]
  {
    long long waves = (long long)(N / 16) * (NCLASS / 16);
    gemm_f32_wmma<<<cdiv(waves * 32, T), T, 0, stream>>>(hprime, W_out, O1, N, NCLASS, NFEAT);
  }

  // output-layer attention
  out_coef<<<cdiv(N, T), T, 0, stream>>>(O1, a_out, esrc2, edst2, N);
  edge_logits<<<cdiv(E, T), T, 0, stream>>>(row, col, esrc2, edst2, ebuf2, emax2, E, N, 1);
  edge_expsum<<<cdiv(E, T), T, 0, stream>>>(row, ebuf2, emax2, denom2, E, N, 1);
  edge_agg_out<<<cdiv((long long)E * 32, T), T, 0, stream>>>(row, col, ebuf2, denom2, O1, outp, E, N);

  // resid = x @ res_W.T  [N,32]
  {
    long long waves = (long long)(N / 16) * (NCLASS / 16);
    gemm_f32_wmma<<<cdiv(waves * 32, T), T, 0, stream>>>(x, WresT, resid, N, NCLASS, NFEAT);
  }

  // relu + residual + bias + layernorm
  final_ln<<<cdiv((long long)N * 32, T), T, 0, stream>>>(outp, resid, res_b, ln_g, ln_b, out, N);
}